// GaussianRasterizer_39857296507437
// MI455X (gfx1250) — compile-verified
//
#include <hip/hip_runtime.h>
#include <math.h>

// ---------------- problem constants ----------------
static constexpr int   NG   = 2048;
static constexpr int   IMH  = 96;
static constexpr int   IMW  = 96;
static constexpr int   NPIX = IMH * IMW;          // 9216
static constexpr int   NSEG = 8;                  // gaussian segments per pixel group
static constexpr int   GSEG = NG / NSEG;          // 256 gaussians per segment
static constexpr float TANX = 0.7f, TANY = 0.7f;
static constexpr float SCALE_MOD = 1.0f;

// SH constants
static constexpr float C0v = 0.28209479177387814f;
static constexpr float C1v = 0.4886025119029199f;
static constexpr float C2_0 = 1.0925484305920792f,  C2_1 = -1.0925484305920792f,
                       C2_2 = 0.31539156525252005f, C2_3 = -1.0925484305920792f,
                       C2_4 = 0.5462742152960396f;
static constexpr float C3_0 = -0.5900435899266435f, C3_1 = 2.890611442640554f,
                       C3_2 = -0.4570457994644658f, C3_3 = 0.3731763325901154f,
                       C3_4 = -0.4570457994644658f, C3_5 = 1.445305721320277f,
                       C3_6 = -0.5900435899266435f;

typedef __attribute__((ext_vector_type(2))) float v2f;
typedef __attribute__((ext_vector_type(8))) float v8f;
typedef int v4i __attribute__((__vector_size__(16)));
typedef __attribute__((address_space(1))) v4i gv4i;   // global 16B vector
typedef __attribute__((address_space(3))) v4i lv4i;   // LDS    16B vector

#if defined(__gfx1250__) && __has_builtin(__builtin_amdgcn_global_load_async_to_lds_b128)
#define HAVE_ASYNC_LDS 1
#else
#define HAVE_ASYNC_LDS 0
#endif

__device__ __forceinline__ float clampf(float v, float lo, float hi) {
    return fminf(fmaxf(v, lo), hi);
}

// ================= Kernel 1: per-gaussian preprocess =================
__global__ __launch_bounds__(256)
void gs_preprocess(const float* __restrict__ means3D,
                   const float* __restrict__ shs,
                   const float* __restrict__ opac,
                   const float* __restrict__ scales,
                   const float* __restrict__ rots,
                   const float* __restrict__ vm,
                   const float* __restrict__ pm,
                   const float* __restrict__ campos,
                   int*   __restrict__ radii_out,
                   float* __restrict__ rec)
{
    int i = blockIdx.x * blockDim.x + threadIdx.x;
    if (i >= NG) return;

    float x = means3D[3*i+0], y = means3D[3*i+1], z = means3D[3*i+2];

    float pv0   = x*vm[0] + y*vm[4] + z*vm[8]  + vm[12];
    float pv1   = x*vm[1] + y*vm[5] + z*vm[9]  + vm[13];
    float depth = x*vm[2] + y*vm[6] + z*vm[10] + vm[14];

    float ph0 = x*pm[0] + y*pm[4] + z*pm[8]  + pm[12];
    float ph1 = x*pm[1] + y*pm[5] + z*pm[9]  + pm[13];
    float ph3 = x*pm[3] + y*pm[7] + z*pm[11] + pm[15];
    float invw = 1.0f / (ph3 + 1e-7f);
    float pr0 = ph0 * invw, pr1 = ph1 * invw;

    float dxc = x - campos[0], dyc = y - campos[1], dzc = z - campos[2];
    float rinv = rsqrtf(dxc*dxc + dyc*dyc + dzc*dzc);
    float sx = dxc*rinv, sy = dyc*rinv, sz = dzc*rinv;
    float xx = sx*sx, yy = sy*sy, zz = sz*sz;
    float xy = sx*sy, yz = sy*sz, xz = sx*sz;

    float b1  = -C1v*sy, b2 = C1v*sz, b3 = -C1v*sx;
    float b4  = C2_0*xy, b5 = C2_1*yz, b6 = C2_2*(2.0f*zz - xx - yy);
    float b7  = C2_3*xz, b8 = C2_4*(xx - yy);
    float b9  = C3_0*sy*(3.0f*xx - yy);
    float b10 = C3_1*xy*sz;
    float b11 = C3_2*sy*(4.0f*zz - xx - yy);
    float b12 = C3_3*sz*(2.0f*zz - 3.0f*xx - 3.0f*yy);
    float b13 = C3_4*sx*(4.0f*zz - xx - yy);
    float b14 = C3_5*sz*(xx - yy);
    float b15 = C3_6*sx*(xx - 3.0f*yy);

    const float* sh = shs + i*48;
    float col[3];
    #pragma unroll
    for (int ch = 0; ch < 3; ++ch) {
        float r = C0v*sh[ch]
                + b1*sh[3+ch]  + b2*sh[6+ch]   + b3*sh[9+ch]
                + b4*sh[12+ch] + b5*sh[15+ch]  + b6*sh[18+ch]
                + b7*sh[21+ch] + b8*sh[24+ch]
                + b9*sh[27+ch] + b10*sh[30+ch] + b11*sh[33+ch]
                + b12*sh[36+ch]+ b13*sh[39+ch] + b14*sh[42+ch] + b15*sh[45+ch];
        col[ch] = fmaxf(r + 0.5f, 0.0f);
    }

    float q0 = rots[4*i+0], q1 = rots[4*i+1], q2 = rots[4*i+2], q3 = rots[4*i+3];
    float qn = rsqrtf(q0*q0 + q1*q1 + q2*q2 + q3*q3);
    float qr = q0*qn, qx = q1*qn, qy = q2*qn, qz = q3*qn;
    float R00 = 1.0f - 2.0f*(qy*qy + qz*qz), R01 = 2.0f*(qx*qy - qr*qz), R02 = 2.0f*(qx*qz + qr*qy);
    float R10 = 2.0f*(qx*qy + qr*qz), R11 = 1.0f - 2.0f*(qx*qx + qz*qz), R12 = 2.0f*(qy*qz - qr*qx);
    float R20 = 2.0f*(qx*qz - qr*qy), R21 = 2.0f*(qy*qz + qr*qx), R22 = 1.0f - 2.0f*(qx*qx + qy*qy);

    float s0 = scales[3*i+0]*SCALE_MOD; s0 *= s0;
    float s1 = scales[3*i+1]*SCALE_MOD; s1 *= s1;
    float s2 = scales[3*i+2]*SCALE_MOD; s2 *= s2;

    float S00 = R00*R00*s0 + R01*R01*s1 + R02*R02*s2;
    float S01 = R00*R10*s0 + R01*R11*s1 + R02*R12*s2;
    float S02 = R00*R20*s0 + R01*R21*s1 + R02*R22*s2;
    float S11 = R10*R10*s0 + R11*R11*s1 + R12*R12*s2;
    float S12 = R10*R20*s0 + R11*R21*s1 + R12*R22*s2;
    float S22 = R20*R20*s0 + R21*R21*s1 + R22*R22*s2;

    float fx = IMW/(2.0f*TANX), fy = IMH/(2.0f*TANY);
    float tz = depth, tz2 = tz*tz;
    float txc = clampf(pv0/tz, -1.3f*TANX, 1.3f*TANX) * tz;
    float tyc = clampf(pv1/tz, -1.3f*TANY, 1.3f*TANY) * tz;
    float J00 = fx/tz, J02 = -fx*txc/tz2;
    float J11 = fy/tz, J12 = -fy*tyc/tz2;

    float M00 = J00*vm[0] + J02*vm[2];
    float M01 = J00*vm[4] + J02*vm[6];
    float M02 = J00*vm[8] + J02*vm[10];
    float M10 = J11*vm[1] + J12*vm[2];
    float M11 = J11*vm[5] + J12*vm[6];
    float M12 = J11*vm[9] + J12*vm[10];

    float T00 = M00*S00 + M01*S01 + M02*S02;
    float T01 = M00*S01 + M01*S11 + M02*S12;
    float T02 = M00*S02 + M01*S12 + M02*S22;
    float T10 = M10*S00 + M11*S01 + M12*S02;
    float T11 = M10*S01 + M11*S11 + M12*S12;
    float T12 = M10*S02 + M11*S12 + M12*S22;
    float a = T00*M00 + T01*M01 + T02*M02 + 0.3f;
    float b = T00*M10 + T01*M11 + T02*M12;
    float c = T10*M10 + T11*M11 + T12*M12 + 0.3f;

    float det = a*c - b*b;
    bool  valid = (depth > 0.2f) && (det > 0.0f);
    float det_s = (det == 0.0f) ? 1.0f : det;
    float conA =  c / det_s;
    float conB = -b / det_s;
    float conC =  a / det_s;

    float mid  = 0.5f*(a + c);
    float lam1 = mid + sqrtf(fmaxf(0.1f, mid*mid - det));
    radii_out[i] = valid ? (int)ceilf(3.0f*sqrtf(fmaxf(lam1, 0.0f))) : 0;

    float px = ((pr0 + 1.0f)*IMW - 1.0f)*0.5f;
    float py = ((pr1 + 1.0f)*IMH - 1.0f)*0.5f;

    float* r = rec + i*12;
    r[0] = conA; r[1] = conB; r[2] = conC;
    r[3] = px;   r[4] = py;   r[5] = depth;
    r[6] = valid ? opac[i] : 0.0f;
    r[7] = valid ? depth : __builtin_inff();     // sort key
    r[8] = col[0]; r[9] = col[1]; r[10] = col[2];
    r[11] = 0.0f;
}

// ================= Kernel 2: stable depth sort (rank counting) =================
__global__ __launch_bounds__(256)
void gs_sort(const float* __restrict__ rec,
             float* __restrict__ coefS,
             float* __restrict__ shadeS,
             float* __restrict__ depthS)
{
    __shared__ float keys[NG];
    int tid = threadIdx.x;
    #pragma unroll
    for (int k = 0; k < NG/256; ++k)
        keys[tid + k*256] = rec[(tid + k*256)*12 + 7];
    __syncthreads();

    int i = blockIdx.x*256 + tid;
    float ki = keys[i];
    int rank = 0;
    for (int j = 0; j < NG; ++j) {
        float kj = keys[j];
        rank += (kj < ki) || ((kj == ki) && (j < i)) ? 1 : 0;
    }

    const float* r = rec + i*12;
    float cA = r[0], cB = r[1], cC = r[2], px = r[3], py = r[4];
    // power(g, pix) = cf . [1, gx, gy, gx^2, gy^2, gx*gy, 0, 0]
    float* cf = coefS + rank*8;
    cf[0] = -0.5f*(cA*px*px + cC*py*py) - cB*px*py;
    cf[1] = cA*px + cB*py;
    cf[2] = cC*py + cB*px;
    cf[3] = -0.5f*cA;
    cf[4] = -0.5f*cC;
    cf[5] = -cB;
    cf[6] = 0.0f;
    cf[7] = 0.0f;

    float4 s; s.x = r[8]; s.y = r[9]; s.z = r[10]; s.w = r[6];
    *(float4*)(shadeS + rank*4) = s;
    depthS[rank] = r[5];
}

// ================= Kernel 3: WMMA rasterize, segmented blend =================
// One block (8 waves) per 16-pixel group. Wave w handles gaussian segment w
// (256 sorted gaussians = 16 chunks of 16). Powers come from two chained
// V_WMMA_F32_16X16X4_F32 per chunk. Per-segment (C,T) partials are composed
// in depth order through LDS (alpha compositing is associative).
__global__ __launch_bounds__(256)
void gs_raster(const float* __restrict__ coefS,
               const float* __restrict__ shadeS,
               const float* __restrict__ depthS,
               const float* __restrict__ bgp,
               float* __restrict__ out)
{
    __shared__ float4 shadeL[NG];           // {r,g,b,op} per sorted gaussian
    __shared__ float  depthL[NG];
    __shared__ float  part[NSEG][16][5];    // per-segment {R,G,B,D,T} per pixel

    int lane = threadIdx.x & 31;
    int w    = threadIdx.x >> 5;            // segment id 0..7
    int pixBase = blockIdx.x * 16;
    int rowY = pixBase / IMW;
    int col0 = pixBase % IMW;
    int n   = lane & 15;
    int sel = lane >> 4;

    // ---- stage this wave's segment of shade/depth into LDS ----
    int segBase = w * GSEG;
#if HAVE_ASYNC_LDS
    #pragma unroll
    for (int k = 0; k < 8; ++k) {           // 32 lanes * 8 = 256 float4
        int idx = segBase + lane*8 + k;
        __builtin_amdgcn_global_load_async_to_lds_b128(
            (gv4i*)(shadeS + idx*4),
            (lv4i*)&shadeL[idx], 0, 0);
    }
    #pragma unroll
    for (int k = 0; k < 2; ++k) {           // 256 floats = 64 b128 transfers
        int j = segBase + (lane*2 + k)*4;
        __builtin_amdgcn_global_load_async_to_lds_b128(
            (gv4i*)(depthS + j),
            (lv4i*)&depthL[j], 0, 0);
    }
    asm volatile("s_wait_asynccnt 0x0" ::: "memory");
#else
    #pragma unroll
    for (int k = 0; k < 8; ++k) {
        int idx = segBase + lane*8 + k;
        shadeL[idx] = *(const float4*)(shadeS + idx*4);
        depthL[idx] = depthS[idx];
    }
#endif

    // ---- B matrix (4x16, K x N): pixel-basis values ----
    float gx = (float)(col0 + n);
    float gy = (float)rowY;
    v2f B1, B2;
    if (sel == 0) { B1[0] = 1.0f;   B1[1] = gx;    B2[0] = gy*gy; B2[1] = gx*gy; }
    else          { B1[0] = gy;     B1[1] = gx*gx; B2[0] = 0.0f;  B2[1] = 0.0f;  }

    float T = 1.0f;
    float accR = 0.f, accG = 0.f, accB = 0.f, accD = 0.f;

    int cBeg = w * (NG/16/NSEG);            // 16 chunks per segment
    int cEnd = cBeg + (NG/16/NSEG);
    for (int c = cBeg; c < cEnd; ++c) {
        int gA = 16*c + n;                  // A rows: gaussian per lane, K pair by sel
        v2f a1 = *(const v2f*)(coefS + gA*8 + 2*sel);
        v2f a2 = *(const v2f*)(coefS + gA*8 + 4 + 2*sel);

        if (c + 1 < cEnd)                   // global_prefetch_b8 of next chunk
            __builtin_prefetch(coefS + (c+1)*16*8, 0, 1);

        v8f acc = {};
        acc = __builtin_amdgcn_wmma_f32_16x16x4_f32(false, a1, false, B1,
                                                    (short)0, acc, false, false);
        acc = __builtin_amdgcn_wmma_f32_16x16x4_f32(false, a2, false, B2,
                                                    (short)0, acc, false, false);

        int gbase = 16*c + 8*sel;
        float al[8], cr[8], cg[8], cb[8], dp[8];
        float p = 1.0f;
        #pragma unroll
        for (int v = 0; v < 8; ++v) {
            float4 s4 = shadeL[gbase + v];  // ds_load_b128 broadcast
            float power = acc[v];
            float e = 0.0f;
            if (power <= 0.0f) {            // exec-mask early-out skips v_exp
                e = fminf(0.99f, s4.w * __expf(power));
                if (e < (1.0f/255.0f)) e = 0.0f;
            }
            al[v] = e;
            cr[v] = s4.x; cg[v] = s4.y; cb[v] = s4.z;
            dp[v] = depthL[gbase + v];
            p *= (1.0f - e);
        }

        float pp = __shfl_xor(p, 16, 32);   // cross-half (1-alpha) product
        float Trun = sel ? T * pp : T;
        #pragma unroll
        for (int v = 0; v < 8; ++v) {
            float ww = al[v] * Trun;
            accR += ww*cr[v]; accG += ww*cg[v]; accB += ww*cb[v]; accD += ww*dp[v];
            Trun *= (1.0f - al[v]);
        }
        T *= p * pp;
    }

    // combine half-wave partials per pixel
    accR += __shfl_xor(accR, 16, 32);
    accG += __shfl_xor(accG, 16, 32);
    accB += __shfl_xor(accB, 16, 32);
    accD += __shfl_xor(accD, 16, 32);

    if (sel == 0) {
        part[w][n][0] = accR; part[w][n][1] = accG;
        part[w][n][2] = accB; part[w][n][3] = accD;
        part[w][n][4] = T;
    }
    __syncthreads();

    // ---- in-order composition of the 8 segments (threads 0..15) ----
    if (threadIdx.x < 16) {
        int t = threadIdx.x;
        float Cr = 0.f, Cg = 0.f, Cb = 0.f, Dd = 0.f, Tt = 1.0f;
        #pragma unroll
        for (int s = 0; s < NSEG; ++s) {
            Cr += Tt * part[s][t][0];
            Cg += Tt * part[s][t][1];
            Cb += Tt * part[s][t][2];
            Dd += Tt * part[s][t][3];
            Tt *= part[s][t][4];
        }
        int pix = pixBase + t;
        float bg0 = bgp[0], bg1 = bgp[1], bg2 = bgp[2];
        out[0*NPIX + pix] = Cr + Tt*bg0;
        out[1*NPIX + pix] = Cg + Tt*bg1;
        out[2*NPIX + pix] = Cb + Tt*bg2;
        out[3*NPIX + NG + pix] = Dd;        // depth image after int32 radii block
    }
}

// ================= host launch =================
extern "C" void kernel_launch(void* const* d_in, const int* in_sizes, int n_in,
                              void* d_out, int out_size, void* d_ws, size_t ws_size,
                              hipStream_t stream)
{
    const float* means3D = (const float*)d_in[0];
    // d_in[1] = means2D (unused by reference)
    const float* shs     = (const float*)d_in[2];
    const float* opac    = (const float*)d_in[3];
    const float* scales  = (const float*)d_in[4];
    const float* rots    = (const float*)d_in[5];
    const float* vm      = (const float*)d_in[6];
    const float* pm      = (const float*)d_in[7];
    const float* campos  = (const float*)d_in[8];
    const float* bg      = (const float*)d_in[9];

    float* out = (float*)d_out;
    float* wsf = (float*)d_ws;
    // ws layout (floats): coefS[2048*8] | shadeS[2048*4] | depthS[2048] | rec[2048*12]
    float* coefS  = wsf;
    float* shadeS = wsf + NG*8;
    float* depthS = wsf + NG*8 + NG*4;
    float* rec    = wsf + NG*8 + NG*4 + NG;
    int*   radii  = (int*)(out + 3*NPIX);

    gs_preprocess<<<NG/256, 256, 0, stream>>>(means3D, shs, opac, scales, rots,
                                              vm, pm, campos, radii, rec);
    gs_sort<<<NG/256, 256, 0, stream>>>(rec, coefS, shadeS, depthS);
    // one block per 16-pixel group; 8 waves = 8 gaussian segments per block
    gs_raster<<<NPIX/16, 256, 0, stream>>>(coefS, shadeS, depthS, bg, out);
}